// KVCache_51891794870282
// MI455X (gfx1250) — compile-verified
//
#include <hip/hip_runtime.h>
#include <hip/hip_bf16.h>

// ---------------------------------------------------------------------------
// KV-cache scatter update for MI455X (gfx1250).
// Pure bandwidth problem: ~578 MiB total traffic -> ~25 us @ 23.3 TB/s.
//   1) inverse map inv[T] in d_ws (inv[pos[s]] = s, else -1)
//   2) fused streaming merge per cache: out row <- chunk row or old cache row
//      (each output byte written once, each source byte read once)
//   3) k/v caches (2 x 128 MiB bf16) move through the CDNA5 async
//      global<->LDS path: 8 async b128 loads in flight per wave,
//      one s_wait_asynccnt, 8 async stores, ENDPGM implicit wait-idle.
//      NT temporal hints keep the 578 MiB stream from thrashing L2.
// ---------------------------------------------------------------------------

#define CB 2
#define CT 8192
#define CH 32
#define CD 128
#define CL 16
#define CS 2048

typedef unsigned int u32x4 __attribute__((ext_vector_type(4)));

// Output byte offsets (outputs concatenated flat in return order, natural dtypes)
#define OFF_K   0ull                                    // k_cache  bf16: 134,217,728 B
#define OFF_V   134217728ull                            // v_cache  bf16: 134,217,728 B
#define OFF_VN  268435456ull                            // v_norm   fp16:   1,048,576 B
#define OFF_KH  269484032ull                            // k_hard   i32 :  33,554,432 B
#define OFF_PRE 303038464ull                            // prefill_len i32

// u32x4 (16 B) element counts
#define KV_U4     8388608u   // CB*CT*CH*CD*2/16 (per cache)
#define VN_U4     65536u     // CB*CT*CH*2/16
#define KH_U4     2097152u   // CB*CT*CH*CL*4/16

// ---------------------------------------------------------------------------
__global__ void init_inv_kernel(int* __restrict__ inv) {
    unsigned i = blockIdx.x * 256u + threadIdx.x;      // grid covers T exactly
    inv[i] = -1;
}

__global__ void fill_inv_kernel(const int* __restrict__ pos, int* __restrict__ inv) {
    unsigned s = blockIdx.x * 256u + threadIdx.x;      // grid covers S exactly
    inv[pos[s]] = (int)s;
}

// ---------------------------------------------------------------------------
// Fused merge for BOTH k and v caches (bf16, row = (b,t,h) = 256 B = 16 u32x4).
// Index math / inv[t] lookup is shared between the two streams. Each block
// handles a contiguous tile of 1024 u32x4 per stream (4 per thread per stream)
// staged through 32 KB of LDS via the gfx1250 async-copy instructions.
// Grid is exact -> EXEC all ones at every async op (ISA requirement met).
__global__ void fuse_kv_kernel(const u32x4* __restrict__ kchunk,  // [B,S,H,D] bf16
                               const u32x4* __restrict__ kcache,  // [B,T,H,D] bf16
                               const u32x4* __restrict__ vchunk,
                               const u32x4* __restrict__ vcache,
                               const int*   __restrict__ inv,
                               u32x4*       __restrict__ kout,
                               u32x4*       __restrict__ vout) {
    __shared__ u32x4 smem[2048];                       // 32 KB: [0,1024)=k, [1024,2048)=v
    const unsigned tid  = threadIdx.x;
    const unsigned base = blockIdx.x * 1024u;

    unsigned long long gk[4], gv[4];
    unsigned           ldk[4], ldv[4], doff[4];

#pragma unroll
    for (int k = 0; k < 4; ++k) {
        const unsigned i   = base + (unsigned)k * 256u + tid;  // u32x4 index
        const unsigned row = i >> 4;                            // (b,t,h)
        const unsigned c   = i & 15u;
        const unsigned b   = row / (CT * CH);
        const unsigned rem = row % (CT * CH);
        const unsigned t   = rem / CH;
        const unsigned h   = rem % CH;
        const int      s   = inv[t];
        const unsigned ci  = ((b * CS + (unsigned)s) * CH + h) * 16u + c;

        gk[k] = (unsigned long long)((s >= 0) ? &kchunk[ci] : &kcache[i]);
        gv[k] = (unsigned long long)((s >= 0) ? &vchunk[ci] : &vcache[i]);
        // Low 32 bits of the generic LDS address == LDS byte offset.
        ldk[k] = (unsigned)(unsigned long long)(&smem[(unsigned)k * 256u + tid]);
        ldv[k] = (unsigned)(unsigned long long)(&smem[1024u + (unsigned)k * 256u + tid]);
        doff[k] = i * 16u;                             // byte offset, < 2^28
    }

    // 8 async global->LDS b128 loads in flight per wave (4 KB/wave), GV mode.
#pragma unroll
    for (int k = 0; k < 4; ++k)
        asm volatile("global_load_async_to_lds_b128 %0, %1, off th:TH_LOAD_NT"
                     :: "v"(ldk[k]), "v"(gk[k]) : "memory");
#pragma unroll
    for (int k = 0; k < 4; ++k)
        asm volatile("global_load_async_to_lds_b128 %0, %1, off th:TH_LOAD_NT"
                     :: "v"(ldv[k]), "v"(gv[k]) : "memory");

    asm volatile("s_wait_asynccnt 0x0" ::: "memory");

    // 8 async LDS->global b128 NT stores (SGPR base + voffset, GVS mode).
    const unsigned long long kb = (unsigned long long)kout;
    const unsigned long long vb = (unsigned long long)vout;
#pragma unroll
    for (int k = 0; k < 4; ++k)
        asm volatile("global_store_async_from_lds_b128 %0, %1, %2 th:TH_STORE_NT"
                     :: "v"(doff[k]), "v"(ldk[k]), "s"(kb) : "memory");
#pragma unroll
    for (int k = 0; k < 4; ++k)
        asm volatile("global_store_async_from_lds_b128 %0, %1, %2 th:TH_STORE_NT"
                     :: "v"(doff[k]), "v"(ldv[k]), "s"(vb) : "memory");
    // S_ENDPGM performs an implicit wait-idle: all async stores complete.
}

// ---------------------------------------------------------------------------
// Fused merge for v_norm cache (fp16, row (b,t) = H*2 = 64 B = 4 x u32x4).
__global__ void fuse_vnorm_kernel(const u32x4* __restrict__ chunk,  // [B,S,H] fp16
                                  const u32x4* __restrict__ cache,  // [B,T,H] fp16
                                  const int*   __restrict__ inv,
                                  u32x4*       __restrict__ out) {
    const unsigned i   = blockIdx.x * 256u + threadIdx.x;  // exact grid (VN_U4)
    const unsigned row = i >> 2;                           // (b,t)
    const unsigned c   = i & 3u;
    const unsigned b   = row / CT;
    const unsigned t   = row % CT;
    const int s = inv[t];
    u32x4 val = (s >= 0)
        ? __builtin_nontemporal_load(&chunk[(b * CS + (unsigned)s) * 4u + c])
        : __builtin_nontemporal_load(&cache[i]);
    __builtin_nontemporal_store(val, &out[i]);
}

// Fused merge for k_hard cache (i32, row (b,t) = H*L*4 = 2048 B = 128 x u32x4).
__global__ void fuse_khard_kernel(const u32x4* __restrict__ chunk,  // [B,S,H,L] i32
                                  const u32x4* __restrict__ cache,  // [B,T,H,L] i32
                                  const int*   __restrict__ inv,
                                  u32x4*       __restrict__ out) {
    const unsigned i   = blockIdx.x * 256u + threadIdx.x;  // exact grid (KH_U4)
    const unsigned row = i >> 7;                           // (b,t)
    const unsigned c   = i & 127u;
    const unsigned b   = row / CT;
    const unsigned t   = row % CT;
    const int s = inv[t];
    u32x4 val = (s >= 0)
        ? __builtin_nontemporal_load(&chunk[(b * CS + (unsigned)s) * 128u + c])
        : __builtin_nontemporal_load(&cache[i]);
    __builtin_nontemporal_store(val, &out[i]);
}

// ---------------------------------------------------------------------------
// prefill_len = max(input_pos) + 1, single-block LDS tree reduction.
__global__ void prefill_max_kernel(const int* __restrict__ pos, int* __restrict__ out) {
    __shared__ int red[256];
    const unsigned tid = threadIdx.x;
    int m = -2147483647 - 1;
    for (unsigned i = tid; i < CS; i += 256u) {
        int p = pos[i];
        m = (p > m) ? p : m;
    }
    red[tid] = m;
    __syncthreads();
    for (unsigned off = 128u; off > 0u; off >>= 1) {
        if (tid < off) {
            int o = red[tid + off];
            if (o > red[tid]) red[tid] = o;
        }
        __syncthreads();
    }
    if (tid == 0) *out = red[0] + 1;
}

// ---------------------------------------------------------------------------
extern "C" void kernel_launch(void* const* d_in, const int* in_sizes, int n_in,
                              void* d_out, int out_size, void* d_ws, size_t ws_size,
                              hipStream_t stream) {
    (void)in_sizes; (void)n_in; (void)out_size; (void)ws_size;

    const int*   pos      = (const int*)  d_in[0];
    const u32x4* k_val    = (const u32x4*)d_in[1];
    const u32x4* v_val    = (const u32x4*)d_in[2];
    const u32x4* vn_val   = (const u32x4*)d_in[3];
    const u32x4* kh_val   = (const u32x4*)d_in[4];
    const u32x4* k_cache  = (const u32x4*)d_in[5];
    const u32x4* v_cache  = (const u32x4*)d_in[6];
    const u32x4* vn_cache = (const u32x4*)d_in[7];
    const u32x4* kh_cache = (const u32x4*)d_in[8];

    char* out = (char*)d_out;
    int*  inv = (int*)d_ws;                            // 8192 * 4 = 32 KB scratch

    // 1) inverse scatter map
    init_inv_kernel<<<CT / 256, 256, 0, stream>>>(inv);
    fill_inv_kernel<<<CS / 256, 256, 0, stream>>>(pos, inv);

    // 2) fused merge passes (single read + single write of each output byte)
    fuse_kv_kernel   <<<KV_U4 / 1024, 256, 0, stream>>>(k_val, k_cache,
                                                        v_val, v_cache, inv,
                                                        (u32x4*)(out + OFF_K),
                                                        (u32x4*)(out + OFF_V));
    fuse_vnorm_kernel<<<VN_U4 / 256, 256, 0, stream>>>(vn_val, vn_cache, inv,
                                                       (u32x4*)(out + OFF_VN));
    fuse_khard_kernel<<<KH_U4 / 256, 256, 0, stream>>>(kh_val, kh_cache, inv,
                                                       (u32x4*)(out + OFF_KH));

    // 3) prefill_len scalar
    prefill_max_kernel<<<1, 256, 0, stream>>>(pos, (int*)(out + OFF_PRE));
}